// CapsuleLayer_65429531787827
// MI455X (gfx1250) — compile-verified
//
#include <hip/hip_runtime.h>

// CapsuleLayer dynamic routing, fully fused for gfx1250 (MI455X).
//
// Algebra: logits_t = pred * Vsum_t elementwise per (k,b,o) (v is broadcast
// over the route-node axis n, logits start at 0), so routing decomposes per
// (k,b,o) channel with softmax over n only. We therefore never materialize
// pred/logits (2x189 MB saved); pred tiles are recomputed each routing pass
// with V_WMMA_F32_16X16X4_F32 and consumed by a flash-style online softmax.
// Working set (x 9.4 MB + W 5.9 MB) is L2-resident on the 192 MB L2.

typedef float v2f __attribute__((ext_vector_type(2)));
typedef float v8f __attribute__((ext_vector_type(8)));

#define KCAP   10
#define NROUTE 1152
#define BATCH  256
#define CIN    8
#define COUT   16
#define NWAVES 8
#define CHUNK  (NROUTE / NWAVES)   // 144 route nodes per wave (divisible by 4)
#define NEG_BIG (-3.0e38f)

__global__ __launch_bounds__(NWAVES * 32, 1)
void caps_route_wmma(const float* __restrict__ x,       // [B, N, CIN]
                     const float* __restrict__ W,       // [K, N, CIN, COUT]
                     const int*   __restrict__ iters_p, // [1]
                     float*       __restrict__ out)     // [K, B, 1, 1, COUT]
{
    // Per-wave partial online-softmax state staged for cross-wave merge.
    __shared__ float sm [NWAVES][32][8];
    __shared__ float sd [NWAVES][32][8];
    __shared__ float sws[NWAVES][32][8];

    const int k    = blockIdx.x / (BATCH / 16);
    const int b0   = (blockIdx.x % (BATCH / 16)) * 16;
    const int lane = threadIdx.x & 31;
    const int wave = threadIdx.x >> 5;
    const int ml   = lane & 15;   // M (A) / col o (B,C,D) within lane half
    const int hi   = lane >> 4;   // lane half selects K-pair (A,B), M+8 (C,D)

    const int iters = iters_p[0];

    // A-matrix 16x4 f32 layout: lane l -> M = l%16; VGPR0/1 hold K = (hi?2:0)/+1.
    // One b64 load of x[b, n, i..i+1] fills {VGPR0, VGPR1} for this lane.
    const float* xbase = x + (size_t)(b0 + ml) * (NROUTE * CIN) + hi * 2;

    // B-matrix 4x16 f32 layout: lane l -> col o = l%16; VGPR0/1 hold rows
    // K = (hi?2:0)/+1. Row stride in W is COUT floats (64 B).
    const float* wbase = W + (((size_t)k * NROUTE) * CIN + hi * 2) * COUT + ml;

    const int n_lo = wave * CHUNK;

    float vsum[8], vlast[8];
#pragma unroll
    for (int r = 0; r < 8; ++r) { vsum[r] = 0.0f; vlast[r] = 0.0f; }

    for (int it = 0; it < iters; ++it) {
        // ---- single fused pass over this wave's route nodes --------------
        float m[8], dd[8], ws[8];
#pragma unroll
        for (int r = 0; r < 8; ++r) { m[r] = NEG_BIG; dd[r] = 0.0f; ws[r] = 0.0f; }

        const float* xp = xbase + (size_t)n_lo * CIN;          // += 8 floats / n
        const float* wp = wbase + (size_t)n_lo * (CIN * COUT); // += 128 floats / n

#pragma unroll 4
        for (int n = 0; n < CHUNK; ++n) {
            // pred tile [16b x 16o] = x[b0:b0+16, n, :] (16x8) @ W[k, n] (8x16)
            v2f a_lo = *(const v2f*)(xp);        // i = {0,1} | {2,3}
            v2f a_hi = *(const v2f*)(xp + 4);    // i = {4,5} | {6,7}
            v2f b_lo; b_lo.x = wp[0];  b_lo.y = wp[16];   // rows i, i+1
            v2f b_hi; b_hi.x = wp[64]; b_hi.y = wp[80];   // rows i+4, i+5

            v8f acc = {};
            acc = __builtin_amdgcn_wmma_f32_16x16x4_f32(
                      false, a_lo, false, b_lo, (short)0, acc, false, false);
            acc = __builtin_amdgcn_wmma_f32_16x16x4_f32(
                      false, a_hi, false, b_hi, (short)0, acc, false, false);

            // online softmax-weighted-sum update, 8 (b,o) cells per lane
#pragma unroll
            for (int r = 0; r < 8; ++r) {
                float p  = acc[r];
                float l  = p * vsum[r];          // logits = pred * Vsum
                float mn = fmaxf(m[r], l);
                float sc = __expf(m[r] - mn);    // rescale old state
                float e  = __expf(l - mn);
                dd[r] = dd[r] * sc + e;
                ws[r] = ws[r] * sc + e * p;
                m[r]  = mn;
            }
            xp += CIN;
            wp += CIN * COUT;
        }

        // ---- merge partial states across the 8 waves ----------------------
#pragma unroll
        for (int r = 0; r < 8; ++r) {
            sm [wave][lane][r] = m[r];
            sd [wave][lane][r] = dd[r];
            sws[wave][lane][r] = ws[r];
        }
        __syncthreads();

#pragma unroll
        for (int r = 0; r < 8; ++r) {
            float M = NEG_BIG;
#pragma unroll
            for (int w = 0; w < NWAVES; ++w) M = fmaxf(M, sm[w][lane][r]);
            float D = 0.0f, WS = 0.0f;
#pragma unroll
            for (int w = 0; w < NWAVES; ++w) {
                float sc = __expf(sm[w][lane][r] - M);
                D  += sd [w][lane][r] * sc;
                WS += sws[w][lane][r] * sc;
            }
            float s  = WS / D;                       // s = Σ_n c_n * pred_n
            float n2 = s * s;                        // squash (dim-2 singleton)
            float v  = (n2 / (1.0f + n2)) * (s / sqrtf(n2));
            vlast[r] = v;
            vsum[r] += v;                            // logits += pred * v
        }
        __syncthreads();  // LDS reusable next pass
    }

    // ---- write v from last iteration: out[k, b, o] -----------------------
    if (wave == 0) {
#pragma unroll
        for (int r = 0; r < 8; ++r) {
            int b_local = r + hi * 8;                // C/D layout: M = r (+8 hi half)
            out[((size_t)k * BATCH + b0 + b_local) * COUT + ml] = vlast[r];
        }
    }
}

extern "C" void kernel_launch(void* const* d_in, const int* in_sizes, int n_in,
                              void* d_out, int out_size, void* d_ws, size_t ws_size,
                              hipStream_t stream) {
    const float* x     = (const float*)d_in[0];   // [256, 1152, 8] f32
    const float* W     = (const float*)d_in[1];   // [10, 1152, 8, 16] f32
    const int*   iters = (const int*)d_in[2];     // scalar (==3)
    float*       out   = (float*)d_out;           // [10, 256, 1, 1, 16] f32

    dim3 grid(KCAP * (BATCH / 16));               // 160 workgroups
    dim3 block(NWAVES * 32);                      // 8 wave32s
    caps_route_wmma<<<grid, block, 0, stream>>>(x, W, iters, out);
}